// GCK3x3Layer_55628416418206
// MI455X (gfx1250) — compile-verified
//
#include <hip/hip_runtime.h>
#include <hip/hip_bf16.h>

typedef float v2f __attribute__((ext_vector_type(2)));
typedef float v8f __attribute__((ext_vector_type(8)));
typedef int   v2i __attribute__((ext_vector_type(2)));

#define C_IN   16
#define C_OUT  32
#define DIM    1024
#define HP     1026          // padded input height/width
#define KTOT   144           // C_IN * 9
#define NSLICE 36            // KTOT / 4  (K-step of V_WMMA_F32_16X16X4_F32)
#define BW     256           // output pixels per block (one row segment)
#define WPAD   146           // padded weight row stride (8B-aligned rows, conflict-free)
#define XW     258           // staged tile width  = BW + 2
#define XROW   (3 * XW)      // 774 floats per input channel

// CDNA5 async global->LDS copy path (no VGPR round-trip, tracked by ASYNCcnt)
#if defined(__has_builtin)
#if __has_builtin(__builtin_amdgcn_global_load_async_to_lds_b64) && \
    __has_builtin(__builtin_amdgcn_s_wait_asynccnt)
#define USE_ASYNC_LDS 1
#endif
#endif

// Builtin expects: (v2i AS(1)* src, v2i AS(3)* dst, imm offset, imm cpol)
#define AS_GLOBAL_B64(p) ((__attribute__((address_space(1))) v2i*)(p))
#define AS_LDS_B64(p)    ((__attribute__((address_space(3))) v2i*)(p))

// out[o,h,w] = sum_{c,i,j} W[o,c,i,j] * x[c,h+i,w+j]
// (exactly the GCK reference: the basis decomposition is an identity transform)
__global__ __launch_bounds__(256)
void gck_conv3x3_wmma_f32(const float* __restrict__ x,
                          const float* __restrict__ w,
                          float* __restrict__ out)
{
    __shared__ __align__(16) float sW[C_OUT * WPAD];   // 18.7 KB
    __shared__ __align__(16) float sX[C_IN * XROW];    // 48.4 KB

    const int tid  = threadIdx.x;
    const int row0 = blockIdx.y;          // output row
    const int col0 = blockIdx.x * BW;     // first output column of this block

#if USE_ASYNC_LDS
    // ---- weights: 32 rows x 72 async-b64 chunks ----
    // global row start = o*144*4 (8B aligned), LDS row start = o*146*4 (8B aligned)
    for (int i = tid; i < C_OUT * (KTOT / 2); i += 256) {
        const int o = i / (KTOT / 2);
        const int p = i - o * (KTOT / 2);
        __builtin_amdgcn_global_load_async_to_lds_b64(
            AS_GLOBAL_B64(w + o * KTOT + p * 2),
            AS_LDS_B64(sW + o * WPAD + p * 2), 0, 0);
    }
    // ---- x strip: 48 rows (16c x 3h) x 129 async-b64 chunks ----
    // global row start byte ≡ 0 or 8 (mod 16) -> always 8B aligned
    for (int i = tid; i < C_IN * 3 * (XW / 2); i += 256) {
        const int rrow = i / (XW / 2);             // 0..47 == c*3 + hh
        const int p    = i - rrow * (XW / 2);
        const int c    = rrow / 3;
        const int hh   = rrow - c * 3;
        __builtin_amdgcn_global_load_async_to_lds_b64(
            AS_GLOBAL_B64(x + (size_t)c * (HP * HP) + (size_t)(row0 + hh) * HP + col0 + p * 2),
            AS_LDS_B64(sX + rrow * XW + p * 2), 0, 0);
    }
    __builtin_amdgcn_s_wait_asynccnt(0);
#else
    // ---- fallback: synchronous staging ----
    for (int i = tid; i < C_OUT * KTOT; i += 256) {
        const int o = i / KTOT;
        const int k = i - o * KTOT;
        sW[o * WPAD + k] = w[i];
    }
    for (int i = tid; i < C_IN * XROW; i += 256) {
        const int c  = i / XROW;
        const int r  = i - c * XROW;
        const int hh = r / XW;
        const int ww = r - hh * XW;
        sX[i] = x[(size_t)c * (HP * HP) + (size_t)(row0 + hh) * HP + (col0 + ww)];
    }
#endif
    __syncthreads();

    const int wave = tid >> 5;
    const int lane = tid & 31;
    const int m    = lane & 15;     // row/col within 16-wide fragment
    const int hv   = lane >> 4;     // half-wave select (K rows 0,1 vs 2,3)

    // A fragment base: lane m holds weight row m (and m+16 for the 2nd M-tile);
    // VGPR0/1 carry K = 2*hv + {0,1} -> fold 2*hv into the base pointer.
    const float* wr0 = &sW[m * WPAD + 2 * hv];
    const float* wr1 = &sW[(m + 16) * WPAD + 2 * hv];
    // B fragment bases: lane m supplies output column (strip_base + m)
    const float* xc0 = &sX[wave * 32 + m];
    const float* xc1 = xc0 + 16;

    v8f acc00 = {}, acc01 = {}, acc10 = {}, acc11 = {};

#pragma unroll
    for (int ks = 0; ks < NSLICE; ++ks) {
        const int k0 = ks * 4;

        v2f a0, a1;
        a0.x = wr0[k0];     a0.y = wr0[k0 + 1];
        a1.x = wr1[k0];     a1.y = wr1[k0 + 1];

        // Load all 4 K-values of this slice at compile-time-constant LDS
        // offsets; each half-wave keeps the pair it owns (v_cndmask).
        float b0v[4], b1v[4];
#pragma unroll
        for (int q = 0; q < 4; ++q) {
            const int k   = k0 + q;
            const int c   = k / 9;
            const int rr  = (k - c * 9) / 3;
            const int jj  = k - c * 9 - rr * 3;
            const int off = c * XROW + rr * XW + jj;   // constant after unroll
            b0v[q] = xc0[off];
            b1v[q] = xc1[off];
        }
        v2f b0, b1;
        b0.x = hv ? b0v[2] : b0v[0];
        b0.y = hv ? b0v[3] : b0v[1];
        b1.x = hv ? b1v[2] : b1v[0];
        b1.y = hv ? b1v[3] : b1v[1];

        acc00 = __builtin_amdgcn_wmma_f32_16x16x4_f32(false, a0, false, b0, (short)0, acc00, false, false);
        acc01 = __builtin_amdgcn_wmma_f32_16x16x4_f32(false, a1, false, b0, (short)0, acc01, false, false);
        acc10 = __builtin_amdgcn_wmma_f32_16x16x4_f32(false, a0, false, b1, (short)0, acc10, false, false);
        acc11 = __builtin_amdgcn_wmma_f32_16x16x4_f32(false, a1, false, b1, (short)0, acc11, false, false);
    }

    // ---- C/D layout: VGPR r holds M = 8*hv + r, N = lane%16 ----
    const int    cc0    = col0 + wave * 32 + m;
    const size_t rowoff = (size_t)row0 * DIM;
#pragma unroll
    for (int r = 0; r < 8; ++r) {
        const int o = hv * 8 + r;
        out[(size_t)o        * (DIM * DIM) + rowoff + cc0]      = acc00[r];
        out[(size_t)(o + 16) * (DIM * DIM) + rowoff + cc0]      = acc01[r];
        out[(size_t)o        * (DIM * DIM) + rowoff + cc0 + 16] = acc10[r];
        out[(size_t)(o + 16) * (DIM * DIM) + rowoff + cc0 + 16] = acc11[r];
    }
}

extern "C" void kernel_launch(void* const* d_in, const int* in_sizes, int n_in,
                              void* d_out, int out_size, void* d_ws, size_t ws_size,
                              hipStream_t stream) {
    const float* x   = (const float*)d_in[0];   // (16, 1026, 1026) f32
    const float* w   = (const float*)d_in[1];   // (32, 16, 3, 3)   f32
    float*       out = (float*)d_out;           // (32, 1024, 1024) f32

    dim3 grid(DIM / BW, DIM);   // (4, 1024)
    gck_conv3x3_wmma_f32<<<grid, 256, 0, stream>>>(x, w, out);
}